// ExplaineeGIN_84482006712598
// MI455X (gfx1250) — compile-verified
//
#include <hip/hip_runtime.h>

typedef __attribute__((ext_vector_type(2))) float v2f;
typedef __attribute__((ext_vector_type(8))) float v8f;

#define N_NODES    50000
#define N_EDGES    800000
#define DIM        128
#define NUM_GRAPHS 512

// ---------------------------------------------------------------- zero fill
__global__ void gin_zero_kernel(float* __restrict__ p, long n) {
    long i = (long)blockIdx.x * blockDim.x + threadIdx.x;
    long stride = (long)gridDim.x * blockDim.x;
    for (; i < n; i += stride) p[i] = 0.0f;
}

// ---------------------------------------------------- edge scatter-add (agg)
// agg[dst[e]][d] += X[src[e]][d]; thread per (edge, dim), L2 float atomics.
__global__ void gin_scatter_kernel(const float* __restrict__ X,
                                   const int* __restrict__ src,
                                   const int* __restrict__ dst,
                                   float* __restrict__ Agg) {
    long total = (long)N_EDGES * DIM;
    long i = (long)blockIdx.x * blockDim.x + threadIdx.x;
    long stride = (long)gridDim.x * blockDim.x;
    for (; i < total; i += stride) {
        int e = (int)(i >> 7);
        int d = (int)(i & (DIM - 1));
        int s = src[e];
        int t = dst[e];
        atomicAdd(&Agg[(long)t * DIM + d], X[(long)s * DIM + d]);
    }
}

// -------------------------------------------- fused (Hin[+Add]) @ W + b, ReLU
// One wave -> one 16x16 tile via V_WMMA_F32_16X16X4_F32 (K stepped by 4).
// Block = 256 threads = 8 waves = the 8 column tiles of the 128-wide output.
__global__ __launch_bounds__(256) void gin_gemm_relu_kernel(
    const float* __restrict__ Hin,
    const float* __restrict__ Add,      // nullptr => no residual add
    const float* __restrict__ W,        // [DIM][DIM] row-major (K rows, N cols)
    const float* __restrict__ bias,     // [DIM]
    float* __restrict__ Out,
    int n_rows) {
    __shared__ float lds[16][DIM + 4];  // stride 132 -> conflict-free frag reads

    int m0 = blockIdx.x * 16;

    // Cooperative A-tile load with fused x+agg add.
    for (int t = threadIdx.x; t < 16 * DIM; t += 256) {
        int r = t >> 7;
        int c = t & (DIM - 1);
        int row = m0 + r;
        float v = 0.0f;
        if (row < n_rows) {
            long g = (long)row * DIM + c;
            v = Hin[g];
            if (Add) v += Add[g];
        }
        lds[r][c] = v;
    }
    __syncthreads();

    int wave = threadIdx.x >> 5;
    int lane = threadIdx.x & 31;
    int col0 = wave * 16;
    int n    = lane & 15;                  // A row index == B/N column index
    int koff = (lane >= 16) ? 2 : 0;       // per 16x16x4 f32 fragment layout

    v8f acc = {};
    for (int k = 0; k < DIM; k += 4) {
        v2f a, b;
        a.x = lds[n][k + koff];
        a.y = lds[n][k + koff + 1];
        b.x = W[(k + koff) * DIM + col0 + n];
        b.y = W[(k + koff + 1) * DIM + col0 + n];
        acc = __builtin_amdgcn_wmma_f32_16x16x4_f32(
            /*neg_a=*/false, a, /*neg_b=*/false, b,
            /*c_mod=*/(short)0, acc, /*reuse_a=*/false, /*reuse_b=*/false);
    }

    float bv = bias[col0 + n];
    int rbase = m0 + ((lane >= 16) ? 8 : 0);   // C/D layout: VGPR i -> M=i / M=i+8
#pragma unroll
    for (int i = 0; i < 8; ++i) {
        int row = rbase + i;
        if (row < n_rows) {
            float v = acc[i] + bv;
            Out[(long)row * DIM + col0 + n] = v > 0.0f ? v : 0.0f;
        }
    }
}

// ------------------------------------------------------- per-graph mean pool
__global__ void gin_pool_scatter_kernel(const float* __restrict__ H,
                                        const int* __restrict__ batch,
                                        float* __restrict__ sums) {
    long total = (long)N_NODES * DIM;
    long i = (long)blockIdx.x * blockDim.x + threadIdx.x;
    long stride = (long)gridDim.x * blockDim.x;
    for (; i < total; i += stride) {
        int nidx = (int)(i >> 7);
        int d = (int)(i & (DIM - 1));
        atomicAdd(&sums[(long)batch[nidx] * DIM + d], H[i]);
    }
}

__global__ void gin_count_kernel(const int* __restrict__ batch,
                                 float* __restrict__ counts) {
    int i = blockIdx.x * blockDim.x + threadIdx.x;
    if (i < N_NODES) atomicAdd(&counts[batch[i]], 1.0f);
}

// --------------------------------------------- head: (sums/cnt) @ Wfc + bfc
__global__ void gin_head_kernel(const float* __restrict__ sums,
                                const float* __restrict__ counts,
                                const float* __restrict__ Wfc,   // [DIM][2]
                                const float* __restrict__ bfc,   // [2]
                                float* __restrict__ out) {       // [NUM_GRAPHS][2]
    __shared__ float e[DIM];
    int g = blockIdx.x;
    int d = threadIdx.x;
    float cnt = counts[g];
    if (cnt < 1.0f) cnt = 1.0f;
    e[d] = sums[(long)g * DIM + d] / cnt;
    __syncthreads();
    if (d < 2) {
        float acc = bfc[d];
        for (int k = 0; k < DIM; ++k) acc += e[k] * Wfc[k * 2 + d];
        out[g * 2 + d] = acc;
    }
}

// --------------------------------------------------------------------------
extern "C" void kernel_launch(void* const* d_in, const int* in_sizes, int n_in,
                              void* d_out, int out_size, void* d_ws, size_t ws_size,
                              hipStream_t stream) {
    const float* x   = (const float*)d_in[0];
    const int*   ei  = (const int*)d_in[1];
    const int*   bat = (const int*)d_in[2];
    const float* W1a = (const float*)d_in[3];
    const float* b1a = (const float*)d_in[4];
    const float* W1b = (const float*)d_in[5];
    const float* b1b = (const float*)d_in[6];
    const float* W2a = (const float*)d_in[7];
    const float* b2a = (const float*)d_in[8];
    const float* W2b = (const float*)d_in[9];
    const float* b2b = (const float*)d_in[10];
    const float* Wfc = (const float*)d_in[11];
    const float* bfc = (const float*)d_in[12];

    const int* src = ei;             // edge_index[0]
    const int* dst = ei + N_EDGES;   // edge_index[1]

    float* ws  = (float*)d_ws;
    const long NF = (long)N_NODES * DIM;
    float* agg    = ws;              // 25.6 MB (also reused as h2)
    float* t      = ws + NF;         // 25.6 MB
    float* h1     = ws + 2 * NF;     // 25.6 MB
    float* sums   = ws + 3 * NF;     // 512*128
    float* counts = sums + (long)NUM_GRAPHS * DIM;
    float* h2     = agg;             // layer-2 MLP output reuses agg

    const dim3 zgrid(2048), blk256(256);
    const dim3 sgrid(16384);
    const int  gemm_blocks = (N_NODES + 15) / 16;   // 3125

    // ---- layer 1
    gin_zero_kernel<<<zgrid, blk256, 0, stream>>>(agg, NF);
    gin_scatter_kernel<<<sgrid, blk256, 0, stream>>>(x, src, dst, agg);
    gin_gemm_relu_kernel<<<gemm_blocks, blk256, 0, stream>>>(x, agg, W1a, b1a, t, N_NODES);
    gin_gemm_relu_kernel<<<gemm_blocks, blk256, 0, stream>>>(t, nullptr, W1b, b1b, h1, N_NODES);

    // ---- layer 2
    gin_zero_kernel<<<zgrid, blk256, 0, stream>>>(agg, NF);
    gin_scatter_kernel<<<sgrid, blk256, 0, stream>>>(h1, src, dst, agg);
    gin_gemm_relu_kernel<<<gemm_blocks, blk256, 0, stream>>>(h1, agg, W2a, b2a, t, N_NODES);
    gin_gemm_relu_kernel<<<gemm_blocks, blk256, 0, stream>>>(t, nullptr, W2b, b2b, h2, N_NODES);

    // ---- mean pool + head
    gin_zero_kernel<<<dim3(64), blk256, 0, stream>>>(sums, (long)NUM_GRAPHS * DIM + NUM_GRAPHS);
    gin_pool_scatter_kernel<<<dim3(8192), blk256, 0, stream>>>(h2, bat, sums);
    gin_count_kernel<<<dim3((N_NODES + 255) / 256), blk256, 0, stream>>>(bat, counts);
    gin_head_kernel<<<NUM_GRAPHS, dim3(DIM), 0, stream>>>(sums, counts, Wfc, bfc, (float*)d_out);
}